// E54DiagonalNoProjCell_47656957116852
// MI455X (gfx1250) — compile-verified
//
#include <hip/hip_runtime.h>
#include <stdint.h>

// ---------------- problem constants ----------------
#define T_DIM 2048
#define B_DIM 8
#define D_DIM 2048
#define CHUNK 64                       // timesteps per chunk (power of two)
#define LOG2_CHUNK 6
#define NCHUNK (T_DIM / CHUNK)         // 32
#define NBUF 16                        // async pipeline depth (LDS stages)
#define TPB 256                        // threads per block
#define DVEC 4                         // floats per thread (float4)
#define DTILE (TPB * DVEC)             // 1024 d-values per block
#define DBLK (D_DIM / DTILE)           // 2 blocks along D

typedef float float4v __attribute__((ext_vector_type(4)));
typedef int   int4v   __attribute__((ext_vector_type(4)));

// ---------------- CDNA5 async global->LDS path ----------------
#if defined(__has_builtin)
#if __has_builtin(__builtin_amdgcn_global_load_async_to_lds_b128)
#define HAVE_ASYNC_LDS 1
#endif
#if __has_builtin(__builtin_amdgcn_s_wait_asynccnt)
#define WAIT_ASYNC(n) __builtin_amdgcn_s_wait_asynccnt(n)
#endif
#endif
#ifndef HAVE_ASYNC_LDS
#define HAVE_ASYNC_LDS 0
#endif
#ifndef WAIT_ASYNC
#define WAIT_ASYNC(n) asm volatile("s_wait_asynccnt %0" ::"i"(n) : "memory")
#endif

typedef __attribute__((address_space(1))) int4v* gptr_b128;  // global int4*
typedef __attribute__((address_space(3))) int4v* lptr_b128;  // LDS int4*

__device__ __forceinline__ void async_load_b128(const float* gsrc, float* lds_dst) {
#if HAVE_ASYNC_LDS
  __builtin_amdgcn_global_load_async_to_lds_b128(
      (gptr_b128)gsrc, (lptr_b128)lds_dst, /*offset=*/0, /*cpol=*/0);
#else
  *(float4v*)lds_dst = *(const float4v*)gsrc;  // sync fallback
#endif
}

__device__ __forceinline__ float fast_sigmoid(float v) {
  // 1 / (1 + e^-v) with v_exp_f32 + v_rcp_f32
  return __builtin_amdgcn_rcpf(1.0f + __expf(-v));
}

// ================= pass 1: per-chunk partial end-state (zero init) =========
__global__ __launch_bounds__(TPB) void pass1_partials(
    const float* __restrict__ x, const float* __restrict__ log_d,
    const float* __restrict__ bias, float* __restrict__ A) {
  __shared__ __align__(16) float sbuf[NBUF * DTILE];  // 64 KB
  const int tid  = threadIdx.x;
  const int d0   = blockIdx.x * DTILE + tid * DVEC;
  const int b    = blockIdx.y;
  const int c    = blockIdx.z;
  const int t0   = c * CHUNK;

  float4v ld = *(const float4v*)(log_d + d0);
  float4v bb = *(const float4v*)(bias  + d0);
  float4v dec;
#pragma unroll
  for (int i = 0; i < DVEC; ++i) dec[i] = 1.0f / (1.0f + __expf(-ld[i]));

  const size_t gstride = (size_t)B_DIM * D_DIM;
  const float* gx = x + ((size_t)t0 * B_DIM + b) * D_DIM + d0;
  float* myslot = &sbuf[tid * DVEC];

#pragma unroll
  for (int s = 0; s < NBUF; ++s)
    async_load_b128(gx + (size_t)s * gstride, myslot + s * DTILE);

  float4v h = {0.f, 0.f, 0.f, 0.f};
  for (int t = 0; t < CHUNK; ++t) {
    WAIT_ASYNC(NBUF - 1);  // oldest stage has landed in LDS
    float4v xv = *(const float4v*)(myslot + (t % NBUF) * DTILE);
#pragma unroll
    for (int i = 0; i < DVEC; ++i)
      h[i] = __builtin_fmaf(dec[i], xv[i] + h[i], bb[i]);
    const int tn = t + NBUF;
    if (tn < CHUNK) {
      asm volatile("s_wait_dscnt 0" ::: "memory");  // slot free before DMA reuse
      async_load_b128(gx + (size_t)tn * gstride, myslot + (tn % NBUF) * DTILE);
    }
  }
  *(float4v*)(A + ((size_t)c * B_DIM + b) * D_DIM + d0) = h;
}

// ================= combine: serial carry over 32 chunks per channel ========
__global__ __launch_bounds__(TPB) void combine_carries(
    const float* __restrict__ h0, const float* __restrict__ log_d,
    const float* __restrict__ A, float* __restrict__ Hin,
    float* __restrict__ h_row0) {
  const int tid = blockIdx.x * blockDim.x + threadIdx.x;  // 0 .. B*D/4-1
  const int b   = tid / (D_DIM / DVEC);
  const int d0  = (tid % (D_DIM / DVEC)) * DVEC;

  float4v ld = *(const float4v*)(log_d + d0);
  float4v dec, dL;
#pragma unroll
  for (int i = 0; i < DVEC; ++i) dec[i] = 1.0f / (1.0f + __expf(-ld[i]));
  dL = dec;
#pragma unroll
  for (int s = 0; s < LOG2_CHUNK; ++s) dL *= dL;  // dec^CHUNK

  float4v h = *(const float4v*)(h0 + (size_t)b * D_DIM + d0);
  *(float4v*)(h_row0 + (size_t)b * D_DIM + d0) = h;  // h[0] = h0

  for (int c = 0; c < NCHUNK; ++c) {
    *(float4v*)(Hin + ((size_t)c * B_DIM + b) * D_DIM + d0) = h;
    float4v a = *(const float4v*)(A + ((size_t)c * B_DIM + b) * D_DIM + d0);
#pragma unroll
    for (int i = 0; i < DVEC; ++i) h[i] = __builtin_fmaf(dL[i], h[i], a[i]);
  }
}

// ================= pass 2: true scan per chunk, write h + silu output ======
__global__ __launch_bounds__(TPB) void pass2_scan(
    const float* __restrict__ x, const float* __restrict__ log_d,
    const float* __restrict__ bias, const float* __restrict__ Hin,
    float* __restrict__ out, float* __restrict__ hout) {
  __shared__ __align__(16) float sbuf[NBUF * DTILE];
  const int tid = threadIdx.x;
  const int d0  = blockIdx.x * DTILE + tid * DVEC;
  const int b   = blockIdx.y;
  const int c   = blockIdx.z;
  const int t0  = c * CHUNK;

  float4v ld = *(const float4v*)(log_d + d0);
  float4v bb = *(const float4v*)(bias  + d0);
  float4v dec;
#pragma unroll
  for (int i = 0; i < DVEC; ++i) dec[i] = 1.0f / (1.0f + __expf(-ld[i]));

  const size_t gstride = (size_t)B_DIM * D_DIM;
  const float* gx = x + ((size_t)t0 * B_DIM + b) * D_DIM + d0;
  float* myslot = &sbuf[tid * DVEC];

#pragma unroll
  for (int s = 0; s < NBUF; ++s)
    async_load_b128(gx + (size_t)s * gstride, myslot + s * DTILE);

  float4v h = *(const float4v*)(Hin + ((size_t)c * B_DIM + b) * D_DIM + d0);

  for (int t = 0; t < CHUNK; ++t) {
    WAIT_ASYNC(NBUF - 1);
    float4v xv = *(const float4v*)(myslot + (t % NBUF) * DTILE);
    float4v o;
#pragma unroll
    for (int i = 0; i < DVEC; ++i) {
      h[i] = __builtin_fmaf(dec[i], xv[i] + h[i], bb[i]);
      o[i] = h[i] * h[i] * fast_sigmoid(h[i]);  // h * silu(h)
    }
    const size_t gt = (size_t)(t0 + t);
    *(float4v*)(hout + ((gt + 1) * B_DIM + b) * D_DIM + d0) = h;  // h[t+1]
    *(float4v*)(out  + (gt       * B_DIM + b) * D_DIM + d0) = o;
    const int tn = t + NBUF;
    if (tn < CHUNK) {
      asm volatile("s_wait_dscnt 0" ::: "memory");
      async_load_b128(gx + (size_t)tn * gstride, myslot + (tn % NBUF) * DTILE);
    }
  }
}

// ================= fallback: single-pass scan (if ws too small) ============
__global__ __launch_bounds__(TPB) void fallback_full_scan(
    const float* __restrict__ x, const float* __restrict__ h0,
    const float* __restrict__ log_d, const float* __restrict__ bias,
    float* __restrict__ out, float* __restrict__ hout) {
  const int tid = blockIdx.x * blockDim.x + threadIdx.x;
  const int b   = tid / (D_DIM / DVEC);
  const int d0  = (tid % (D_DIM / DVEC)) * DVEC;

  float4v ld = *(const float4v*)(log_d + d0);
  float4v bb = *(const float4v*)(bias  + d0);
  float4v dec;
#pragma unroll
  for (int i = 0; i < DVEC; ++i) dec[i] = 1.0f / (1.0f + __expf(-ld[i]));

  float4v h = *(const float4v*)(h0 + (size_t)b * D_DIM + d0);
  *(float4v*)(hout + (size_t)b * D_DIM + d0) = h;

  const size_t gstride = (size_t)B_DIM * D_DIM;
  const float* gx = x + (size_t)b * D_DIM + d0;
  for (int t = 0; t < T_DIM; ++t) {
    float4v xv = *(const float4v*)(gx + (size_t)t * gstride);
    float4v o;
#pragma unroll
    for (int i = 0; i < DVEC; ++i) {
      h[i] = __builtin_fmaf(dec[i], xv[i] + h[i], bb[i]);
      o[i] = h[i] * h[i] * fast_sigmoid(h[i]);
    }
    *(float4v*)(hout + ((size_t)(t + 1) * B_DIM + b) * D_DIM + d0) = h;
    *(float4v*)(out  + ((size_t)t       * B_DIM + b) * D_DIM + d0) = o;
  }
}

// ================= launcher ================================================
extern "C" void kernel_launch(void* const* d_in, const int* in_sizes, int n_in,
                              void* d_out, int out_size, void* d_ws, size_t ws_size,
                              hipStream_t stream) {
  const float* x     = (const float*)d_in[0];  // [T,B,D]
  const float* h0    = (const float*)d_in[1];  // [B,D]
  const float* log_d = (const float*)d_in[2];  // [D]
  const float* bias  = (const float*)d_in[3];  // [D]
  (void)in_sizes; (void)n_in; (void)out_size;

  float* out  = (float*)d_out;                         // [T,B,D]
  float* hout = out + (size_t)T_DIM * B_DIM * D_DIM;   // [T+1,B,D]

  const size_t per   = (size_t)NCHUNK * B_DIM * D_DIM; // floats
  const size_t need  = 2 * per * sizeof(float);        // A + Hin (4 MiB)

  if (ws_size >= need) {
    float* A   = (float*)d_ws;
    float* Hin = A + per;
    dim3 grid(DBLK, B_DIM, NCHUNK), block(TPB);
    pass1_partials<<<grid, block, 0, stream>>>(x, log_d, bias, A);
    combine_carries<<<(B_DIM * D_DIM / DVEC) / TPB, TPB, 0, stream>>>(
        h0, log_d, A, Hin, hout);
    pass2_scan<<<grid, block, 0, stream>>>(x, log_d, bias, Hin, out, hout);
  } else {
    fallback_full_scan<<<(B_DIM * D_DIM / DVEC) / TPB, TPB, 0, stream>>>(
        x, h0, log_d, bias, out, hout);
  }
}